// EcoNENN_31542239822516
// MI455X (gfx1250) — compile-verified
//
#include <hip/hip_runtime.h>
#include <math.h>

#define N_NODES 30000
#define N_EDGES 300000
#define N_E2    330000      // edges + self loops
#define F_NODE  32
#define F_EDGE  16
#define HID     64
#define HEADS   4
#define HC      256         // HEADS * 64
#define NEG_ATT 0.2f
#define NEG_ACT 0.01f
#define LN_EPS  1e-5f
#define BS_STRIDE 72        // LDS row stride: (2*72)%64==16 -> halves hit disjoint banks

typedef __attribute__((ext_vector_type(2))) float v2f;
typedef __attribute__((ext_vector_type(8))) float v8f;

__device__ __forceinline__ float lrelu(float x, float s) { return x > 0.f ? x : s * x; }

__device__ __forceinline__ void atomicMaxF(float* addr, float val) {
    unsigned int* ua = (unsigned int*)addr;
    unsigned int old = *ua;
    while (__uint_as_float(old) < val) {
        unsigned int assumed = old;
        old = atomicCAS(ua, assumed, __float_as_uint(val));
        if (old == assumed) break;
    }
}

__device__ __forceinline__ int edge_src(const int* src, int e2) {
    return e2 < N_EDGES ? src[e2] : (e2 - N_EDGES);
}
__device__ __forceinline__ int edge_dst(const int* dst, int e2) {
    return e2 < N_EDGES ? dst[e2] : (e2 - N_EDGES);
}

// ---------------------------------------------------------------------------
// f32 WMMA GEMM: C[M,N] = act(A[M,K] @ B[K,N] + bias)
// Workgroup = 8 waves sharing one 64-col B strip staged in LDS (stride-72
// padding -> conflict-free ds_load_b32). Each wave computes a 16x64 strip of
// C via 4x V_WMMA_F32_16X16X4_F32 per k-step, A-fragment reused 4x.
// Requires M%16==0, N%64==0, K%4==0, K<=64. EXEC stays all-1s for WMMA.
// ---------------------------------------------------------------------------
__global__ __launch_bounds__(256)
void k_gemm_wmma(const float* __restrict__ A, const float* __restrict__ B,
                 const float* __restrict__ bias, float* __restrict__ C,
                 int M, int K, int N, int relu_act)
{
    __shared__ float Bs[64 * BS_STRIDE];
    int mTiles  = M >> 4;
    int nStrips = N >> 6;
    int ns      = blockIdx.x % nStrips;
    int mtBlk   = blockIdx.x / nStrips;

    // cooperative stage of the K x 64 B panel into LDS
    for (int idx = threadIdx.x; idx < K * 64; idx += 256) {
        int row = idx >> 6, col = idx & 63;
        Bs[row * BS_STRIDE + col] = B[(size_t)row * N + ns * 64 + col];
    }
    __syncthreads();

    int mt = mtBlk * 8 + (threadIdx.x >> 5);
    if (mt >= mTiles) return;              // wave-uniform: EXEC stays all-1s
    int lane = threadIdx.x & 31;
    int r    = lane & 15;
    int half = lane >> 4;

    const float* Arow = A + (size_t)(mt * 16 + r) * K;

    v8f acc0 = {0.f,0.f,0.f,0.f,0.f,0.f,0.f,0.f};
    v8f acc1 = acc0, acc2 = acc0, acc3 = acc0;

    for (int kk = 0; kk < K; kk += 4) {
        v2f a;
        a.x = Arow[kk + 2*half];
        a.y = Arow[kk + 2*half + 1];
        const float* Brow0 = &Bs[(kk + 2*half) * BS_STRIDE + r];
        const float* Brow1 = Brow0 + BS_STRIDE;
        v2f b0; b0.x = Brow0[0];  b0.y = Brow1[0];
        v2f b1; b1.x = Brow0[16]; b1.y = Brow1[16];
        v2f b2; b2.x = Brow0[32]; b2.y = Brow1[32];
        v2f b3; b3.x = Brow0[48]; b3.y = Brow1[48];
        __builtin_prefetch(Arow + kk + 8, 0, 3);
        acc0 = __builtin_amdgcn_wmma_f32_16x16x4_f32(false, a, false, b0, (short)0, acc0, false, false);
        acc1 = __builtin_amdgcn_wmma_f32_16x16x4_f32(false, a, false, b1, (short)0, acc1, false, false);
        acc2 = __builtin_amdgcn_wmma_f32_16x16x4_f32(false, a, false, b2, (short)0, acc2, false, false);
        acc3 = __builtin_amdgcn_wmma_f32_16x16x4_f32(false, a, false, b3, (short)0, acc3, false, false);
    }

    int rowBase = mt * 16 + half * 8;
    #pragma unroll
    for (int j = 0; j < 4; ++j) {
        v8f acc = (j == 0) ? acc0 : (j == 1) ? acc1 : (j == 2) ? acc2 : acc3;
        int col = ns * 64 + j * 16 + r;
        float bv = bias ? bias[col] : 0.f;
        #pragma unroll
        for (int v = 0; v < 8; ++v) {
            float x = acc[v] + bv;
            if (relu_act) x = fmaxf(x, 0.f);
            C[(size_t)(rowBase + v) * N + col] = x;
        }
    }
}

// ---------------------------------------------------------------------------
// Self-loop edge_attr = mean of incoming encoded edge_attr
// ---------------------------------------------------------------------------
__global__ void k_fill(float* __restrict__ p, float v, int n) {
    int t = blockIdx.x * blockDim.x + threadIdx.x;
    if (t < n) p[t] = v;
}

__global__ void k_scatter_mean(const float* __restrict__ ea, const int* __restrict__ dst,
                               float* __restrict__ mea, float* __restrict__ deg)
{
    int t = blockIdx.x * blockDim.x + threadIdx.x;
    if (t >= N_EDGES * HID) return;
    int e = t >> 6, c = t & 63;
    int d = dst[e];
    atomicAdd(mea + (size_t)d * HID + c, ea[t]);
    if (c == 0) atomicAdd(deg + d, 1.0f);
}

__global__ void k_mean_div(float* __restrict__ mea, const float* __restrict__ deg)
{
    int t = blockIdx.x * blockDim.x + threadIdx.x;
    if (t >= N_NODES * HID) return;
    mea[t] /= fmaxf(deg[t >> 6], 1.0f);
}

// wed[k*4+h] = sum_c lin_edge[k, h*64+c] * att_edge[h*64+c]   (fold e @ att_edge)
__global__ void k_fold(const float* __restrict__ line_w, const float* __restrict__ ae,
                       float* __restrict__ wed)
{
    int t = threadIdx.x;          // 256 threads: k=t>>2, h=t&3
    if (t >= HID * HEADS) return;
    int k = t >> 2, h = t & 3;
    float s = 0.f;
    for (int c = 0; c < 64; ++c)
        s += line_w[(size_t)k * HC + h * 64 + c] * ae[h * 64 + c];
    wed[k * HEADS + h] = s;
}

// a_src[n,h] = sum_c h[n,h,c]*att_src[h,c] ; same for a_dst
__global__ void k_node_att(const float* __restrict__ hbuf, const float* __restrict__ as_,
                           const float* __restrict__ ad_, float* __restrict__ asr,
                           float* __restrict__ ads)
{
    int t = blockIdx.x * blockDim.x + threadIdx.x;
    if (t >= N_NODES * HEADS) return;
    int n = t >> 2, h = t & 3;
    const float* row = hbuf + (size_t)n * HC + h * 64;
    const float* wa = as_ + h * 64;
    const float* wd = ad_ + h * 64;
    float s = 0.f, d = 0.f;
    for (int c = 0; c < 64; ++c) { float v = row[c]; s += v * wa[c]; d += v * wd[c]; }
    asr[t] = s; ads[t] = d;
}

// per edge: a = leaky_relu(a_src[s]+a_dst[d]+ea_row@wed, 0.2); segment-max into mmx
__global__ void k_edge_att(const float* __restrict__ ea, const float* __restrict__ mea,
                           const int* __restrict__ src, const int* __restrict__ dst,
                           const float* __restrict__ asr, const float* __restrict__ ads,
                           const float* __restrict__ wed, float* __restrict__ aex,
                           float* __restrict__ mmx)
{
    int e2 = blockIdx.x * blockDim.x + threadIdx.x;
    if (e2 >= N_E2) return;
    int s = edge_src(src, e2), d = edge_dst(dst, e2);
    const float* er = (e2 < N_EDGES) ? ea + (size_t)e2 * HID
                                     : mea + (size_t)(e2 - N_EDGES) * HID;
    float acc[HEADS] = {0.f, 0.f, 0.f, 0.f};
    for (int k = 0; k < HID; ++k) {
        float v = er[k];
        #pragma unroll
        for (int h = 0; h < HEADS; ++h) acc[h] += v * wed[k * HEADS + h];
    }
    #pragma unroll
    for (int h = 0; h < HEADS; ++h) {
        float a = lrelu(asr[s * HEADS + h] + ads[d * HEADS + h] + acc[h], NEG_ATT);
        aex[(size_t)e2 * HEADS + h] = a;
        atomicMaxF(mmx + d * HEADS + h, a);
    }
}

__global__ void k_edge_expsum(float* __restrict__ aex, const int* __restrict__ dst,
                              const float* __restrict__ mmx, float* __restrict__ den)
{
    int t = blockIdx.x * blockDim.x + threadIdx.x;
    if (t >= N_E2 * HEADS) return;
    int e2 = t >> 2, h = t & 3;
    int d = edge_dst(dst, e2);
    float ex = expf(aex[t] - mmx[d * HEADS + h]);
    aex[t] = ex;
    atomicAdd(den + d * HEADS + h, ex);
}

__global__ void k_edge_aggregate(const float* __restrict__ aex, const int* __restrict__ src,
                                 const int* __restrict__ dst, const float* __restrict__ den,
                                 const float* __restrict__ hbuf, float* __restrict__ agg,
                                 float* __restrict__ out_alpha)
{
    int t = blockIdx.x * blockDim.x + threadIdx.x;
    if (t >= N_E2 * HEADS) return;
    int e2 = t >> 2, h = t & 3;
    int s = edge_src(src, e2), d = edge_dst(dst, e2);
    float alpha = aex[t] / den[d * HEADS + h];
    out_alpha[t] = alpha;
    const float* hs = hbuf + (size_t)s * HC + h * 64;
    float* ag = agg + (size_t)d * HC + h * 64;
    for (int c = 0; c < 64; ++c) atomicAdd(ag + c, hs[c] * alpha);
}

// head-mean + bias, leaky_relu(0.01), LayerNorm
__global__ void k_finalize(const float* __restrict__ agg, const float* __restrict__ bias,
                           const float* __restrict__ g, const float* __restrict__ bt,
                           float* __restrict__ xout)
{
    int n = blockIdx.x * blockDim.x + threadIdx.x;
    if (n >= N_NODES) return;
    const float* a = agg + (size_t)n * HC;
    float v[HID];
    float mu = 0.f;
    for (int c = 0; c < HID; ++c) {
        float x = 0.25f * (a[c] + a[64 + c] + a[128 + c] + a[192 + c]) + bias[c];
        x = lrelu(x, NEG_ACT);
        v[c] = x; mu += x;
    }
    mu *= (1.0f / HID);
    float var = 0.f;
    for (int c = 0; c < HID; ++c) { float dv = v[c] - mu; var += dv * dv; }
    var *= (1.0f / HID);
    float inv = rsqrtf(var + LN_EPS);
    for (int c = 0; c < HID; ++c)
        xout[(size_t)n * HID + c] = (v[c] - mu) * inv * g[c] + bt[c];
}

// ---------------------------------------------------------------------------
extern "C" void kernel_launch(void* const* d_in, const int* in_sizes, int n_in,
                              void* d_out, int out_size, void* d_ws, size_t ws_size,
                              hipStream_t stream)
{
    const float* x     = (const float*)d_in[0];
    const int*   ei    = (const int*)d_in[1];
    const float* eattr = (const float*)d_in[2];
    const float* wn    = (const float*)d_in[3];
    const float* bn    = (const float*)d_in[4];
    const float* we    = (const float*)d_in[5];
    const float* be    = (const float*)d_in[6];
    const float* lin[2]  = {(const float*)d_in[7],  (const float*)d_in[15]};
    const float* line[2] = {(const float*)d_in[8],  (const float*)d_in[16]};
    const float* as_[2]  = {(const float*)d_in[9],  (const float*)d_in[17]};
    const float* ad_[2]  = {(const float*)d_in[10], (const float*)d_in[18]};
    const float* ae_[2]  = {(const float*)d_in[11], (const float*)d_in[19]};
    const float* b_[2]   = {(const float*)d_in[12], (const float*)d_in[20]};
    const float* g_[2]   = {(const float*)d_in[13], (const float*)d_in[21]};
    const float* bt_[2]  = {(const float*)d_in[14], (const float*)d_in[22]};
    (void)in_sizes; (void)n_in; (void)out_size; (void)ws_size;

    const int* src = ei;
    const int* dst = ei + N_EDGES;

    float* ws = (float*)d_ws;
    float* x1   = ws;  ws += (size_t)N_NODES * HID;    // encoded nodes
    float* ea   = ws;  ws += (size_t)N_EDGES * HID;    // encoded edges
    float* mea  = ws;  ws += (size_t)N_NODES * HID;    // self-loop mean edge attr
    float* deg  = ws;  ws += N_NODES;
    float* hbuf = ws;  ws += (size_t)N_NODES * HC;     // h = x @ lin
    float* asr  = ws;  ws += N_NODES * HEADS;
    float* ads  = ws;  ws += N_NODES * HEADS;
    float* aex  = ws;  ws += (size_t)N_E2 * HEADS;     // a / exp(a-m) scratch
    float* mmx  = ws;  ws += N_NODES * HEADS;
    float* den  = ws;  ws += N_NODES * HEADS;
    float* agg  = ws;  ws += (size_t)N_NODES * HC;
    float* hln  = ws;  ws += (size_t)N_NODES * HID;    // layer-1 output
    float* wed  = ws;  ws += HID * HEADS;              // folded edge-att weights

    float* out_nodes  = (float*)d_out;
    float* out_alpha1 = out_nodes + (size_t)N_NODES * HID;
    float* out_alpha2 = out_alpha1 + (size_t)N_E2 * HEADS;

    auto nb = [](long long n, int b) { return (unsigned)((n + b - 1) / b); };
    auto gemm = [&](const float* A, const float* B, const float* bias, float* C,
                    int M, int K, int N, int act) {
        int mTiles = M / 16, nStrips = N / 64;
        unsigned blocks = (unsigned)(((mTiles + 7) / 8) * nStrips);
        k_gemm_wmma<<<blocks, 256, 0, stream>>>(A, B, bias, C, M, K, N, act);
    };

    // ---- encoders (WMMA GEMMs) ----
    gemm(x,     wn, bn, x1, N_NODES, F_NODE, HID, 1);
    gemm(eattr, we, be, ea, N_EDGES, F_EDGE, HID, 1);

    // ---- self-loop mean edge attrs ----
    k_fill<<<nb((long long)N_NODES * HID, 256), 256, 0, stream>>>(mea, 0.f, N_NODES * HID);
    k_fill<<<nb(N_NODES, 256), 256, 0, stream>>>(deg, 0.f, N_NODES);
    k_scatter_mean<<<nb((long long)N_EDGES * HID, 256), 256, 0, stream>>>(ea, dst, mea, deg);
    k_mean_div<<<nb((long long)N_NODES * HID, 256), 256, 0, stream>>>(mea, deg);

    // ---- two GAT layers ----
    for (int l = 0; l < 2; ++l) {
        const float* xin = (l == 0) ? x1 : hln;
        float* xout      = (l == 0) ? hln : out_nodes;
        float* out_alpha = (l == 0) ? out_alpha1 : out_alpha2;

        gemm(xin, lin[l], nullptr, hbuf, N_NODES, HID, HC, 0);
        k_fold<<<1, 256, 0, stream>>>(line[l], ae_[l], wed);
        k_node_att<<<nb((long long)N_NODES * HEADS, 256), 256, 0, stream>>>(hbuf, as_[l], ad_[l], asr, ads);

        k_fill<<<nb((long long)N_NODES * HEADS, 256), 256, 0, stream>>>(mmx, -INFINITY, N_NODES * HEADS);
        k_fill<<<nb((long long)N_NODES * HEADS, 256), 256, 0, stream>>>(den, 0.f, N_NODES * HEADS);
        k_fill<<<nb((long long)N_NODES * HC, 256), 256, 0, stream>>>(agg, 0.f, N_NODES * HC);

        k_edge_att<<<nb(N_E2, 256), 256, 0, stream>>>(ea, mea, src, dst, asr, ads, wed, aex, mmx);
        k_edge_expsum<<<nb((long long)N_E2 * HEADS, 256), 256, 0, stream>>>(aex, dst, mmx, den);
        k_edge_aggregate<<<nb((long long)N_E2 * HEADS, 256), 256, 0, stream>>>(aex, src, dst, den,
                                                                               hbuf, agg, out_alpha);
        k_finalize<<<nb(N_NODES, 128), 128, 0, stream>>>(agg, b_[l], g_[l], bt_[l], xout);
    }
}